// SSMLayer_5308579578510
// MI455X (gfx1250) — compile-verified
//
#include <hip/hip_runtime.h>

// ---------------------------------------------------------------------------
// S4 SSM layer on MI455X (gfx1250, wave32) via hierarchical chunked scan.
// All heavy math is fp32 WMMA (v_wmma_f32_16x16x4_f32), K-dim = N = 64.
// Workspace requirement: ~33.6 MiB (see OFF_* map below).
// ---------------------------------------------------------------------------

typedef __attribute__((ext_vector_type(2))) float v2f;
typedef __attribute__((ext_vector_type(8))) float v8f;

#define NST  64
#define LSEQ 4194304
#define NSUB 65536      // LSEQ / 64 sub-chunks
#define NGRP 1024       // NSUB / 64 groups

// float offsets into workspace
#define OFF_AB64   0                      // Ab^64                [64*64]
#define OFF_AB4096 4096                   // Ab^4096              [64*64]
#define OFF_ZT     8192                   // Z^T: ZT[i][n]=(Ab^{63-i}Bb)[n]
#define OFF_TT     12288                  // T64^T: TT[i][k]=K0[k-i] (k>=i)
#define OFF_R1T    16384                  // R1T[n][k]=(Cb Ab^{k+1})[n]
#define OFF_P64T   20480                  // (Ab64^i)^T, i=0..63  [64*4096]
#define OFF_V      282624                 // v[s]   [NSUB*64]
#define OFF_XIN    4476928                // xin[s] [NSUB*64]
#define OFF_E      8671232                // e[g]   [NGRP*64]
#define OFF_SBLK   8736768                // SBLK[g] entering state [NGRP*64]
// total 8802304 floats = 33.6 MiB

__device__ __forceinline__ v8f wmma4(v2f a, v2f b, v8f c) {
  // D = A(16x4 f32) * B(4x16 f32) + C(16x16 f32), wave32, EXEC all-ones here.
  return __builtin_amdgcn_wmma_f32_16x16x4_f32(false, a, false, b, (short)0, c,
                                               false, false);
}

// ---------------------------------------------------------------------------
// Setup: bilinear discretization + small tables (single block, VALU/LDS).
// ---------------------------------------------------------------------------
__global__ __launch_bounds__(256) void k_setup(const float* __restrict__ A,
                                               const float* __restrict__ B,
                                               const float* __restrict__ C,
                                               const float* __restrict__ lstep,
                                               float* __restrict__ ws)
{
  __shared__ float Mg[NST][NST + 1];
  __shared__ float Inv[NST][NST + 1];
  __shared__ float Abl[NST][NST + 1];
  __shared__ float cvec[NST], tvec[NST], K0[NST], Cbs[NST], Bbs[NST];
  __shared__ float piv;
  const int tid = threadIdx.x;
  const float h  = expf(lstep[0]);
  const float hh = 0.5f * h;

  for (int idx = tid; idx < NST * NST; idx += 256) {
    int r = idx >> 6, c = idx & 63;
    float diag = (r == c) ? 1.0f : 0.0f;
    Mg[r][c]  = diag - hh * A[idx];   // I - (h/2)A
    Inv[r][c] = diag;
  }
  __syncthreads();

  // Gauss-Jordan inverse (I-(h/2)A is diagonally dominant; no pivoting)
  for (int p = 0; p < NST; ++p) {
    if (tid == 0) piv = 1.0f / Mg[p][p];
    __syncthreads();
    if (tid < 64)       Mg[p][tid] *= piv;
    else if (tid < 128) Inv[p][tid - 64] *= piv;
    __syncthreads();
    int r = tid >> 2, q = tid & 3;
    float f = Mg[r][p];
    __syncthreads();
    if (r != p) {
      for (int j = 0; j < 16; ++j) {
        int c = q * 16 + j;
        Mg[r][c]  -= f * Mg[p][c];
        Inv[r][c] -= f * Inv[p][c];
      }
    }
    __syncthreads();
  }

  // Mg := I + (h/2)A ; Abl = Inv * Mg  (bilinear Ab)
  for (int idx = tid; idx < NST * NST; idx += 256) {
    int r = idx >> 6, c = idx & 63;
    Mg[r][c] = ((r == c) ? 1.0f : 0.0f) + hh * A[idx];
  }
  __syncthreads();
  for (int o = tid * 16; o < tid * 16 + 16; ++o) {
    int r = o >> 6, c = o & 63;
    float s = 0.f;
    for (int m = 0; m < NST; ++m) s += Inv[r][m] * Mg[m][c];
    Abl[r][c] = s;
  }
  if (tid < 64) {
    float s = 0.f;
    for (int m = 0; m < NST; ++m) s += Inv[tid][m] * B[m];
    Bbs[tid] = h * s;                 // Bb = step * BL @ B
    Cbs[tid] = C[tid];
  }
  __syncthreads();

  // Z columns + K0 taps:  c = Bb; loop: K0[d]=Cb.c; ZT[63-d]=c; c = Ab c
  if (tid < 64) cvec[tid] = Bbs[tid];
  __syncthreads();
  for (int d = 0; d < NST; ++d) {
    if (tid == 0) {
      float s = 0.f;
      for (int m = 0; m < NST; ++m) s += Cbs[m] * cvec[m];
      K0[d] = s;
    }
    if (tid < 64) {
      ws[OFF_ZT + (63 - d) * 64 + tid] = cvec[tid];
      float s = 0.f;
      for (int m = 0; m < NST; ++m) s += Abl[tid][m] * cvec[m];
      tvec[tid] = s;
    }
    __syncthreads();
    if (tid < 64) cvec[tid] = tvec[tid];
    __syncthreads();
  }
  // lower-triangular Toeplitz, stored pre-transposed for the WMMA B operand
  for (int idx = tid; idx < NST * NST; idx += 256) {
    int i = idx >> 6, k = idx & 63;
    ws[OFF_TT + idx] = (k >= i) ? K0[k - i] : 0.0f;
  }
  // R1T[n][k] = (Cb Ab^{k+1})[n] via row recurrence
  if (tid < 64) cvec[tid] = Cbs[tid];
  __syncthreads();
  for (int k = 0; k < NST; ++k) {
    if (tid < 64) {
      float s = 0.f;
      for (int m = 0; m < NST; ++m) s += cvec[m] * Abl[m][tid];
      tvec[tid] = s;
    }
    __syncthreads();
    if (tid < 64) {
      cvec[tid] = tvec[tid];
      ws[OFF_R1T + tid * 64 + k] = tvec[tid];
    }
    __syncthreads();
  }
  // Ab64 = Ab^64 by 6 squarings (Abl consumed)
  for (int itn = 0; itn < 6; ++itn) {
    for (int o = tid * 16; o < tid * 16 + 16; ++o) {
      int r = o >> 6, c = o & 63;
      float s = 0.f;
      for (int m = 0; m < NST; ++m) s += Abl[r][m] * Abl[m][c];
      Inv[r][c] = s;
    }
    __syncthreads();
    for (int idx = tid; idx < NST * NST; idx += 256)
      Abl[idx >> 6][idx & 63] = Inv[idx >> 6][idx & 63];
    __syncthreads();
  }
  for (int idx = tid; idx < NST * NST; idx += 256)
    ws[OFF_AB64 + idx] = Abl[idx >> 6][idx & 63];
}

// ---------------------------------------------------------------------------
// Powers: P64T[i] = (Ab64^i)^T for i=0..63, and Ab4096 = Ab64^64.
// ---------------------------------------------------------------------------
__global__ __launch_bounds__(256) void k_powers(float* __restrict__ ws)
{
  __shared__ float A64[NST][NST + 1];
  __shared__ float cur[NST][NST + 1];
  __shared__ float nxt[NST][NST + 1];
  const int tid = threadIdx.x;
  for (int idx = tid; idx < NST * NST; idx += 256) {
    int r = idx >> 6, c = idx & 63;
    A64[r][c] = ws[OFF_AB64 + idx];
    float id = (r == c) ? 1.0f : 0.0f;
    cur[r][c] = id;
    ws[OFF_P64T + idx] = id;          // (Ab64^0)^T = I
  }
  __syncthreads();
  for (int i = 1; i < NST; ++i) {
    for (int o = tid * 16; o < tid * 16 + 16; ++o) {
      int r = o >> 6, c = o & 63;
      float s = 0.f;
      for (int m = 0; m < NST; ++m) s += cur[r][m] * A64[m][c];
      nxt[r][c] = s;
    }
    __syncthreads();
    for (int idx = tid; idx < NST * NST; idx += 256) {
      int r = idx >> 6, c = idx & 63;
      float v = nxt[r][c];
      cur[r][c] = v;
      ws[OFF_P64T + i * 4096 + c * 64 + r] = v;   // store transposed
    }
    __syncthreads();
  }
  for (int o = tid * 16; o < tid * 16 + 16; ++o) { // Ab4096 = Ab64^63 * Ab64
    int r = o >> 6, c = o & 63;
    float s = 0.f;
    for (int m = 0; m < NST; ++m) s += cur[r][m] * A64[m][c];
    ws[OFF_AB4096 + r * 64 + c] = s;
  }
}

// ---------------------------------------------------------------------------
// Pass 1 (WMMA): per 64-step sub-chunk s:
//   V[s]      = Usub[s] @ Z^T       (state contribution)
//   Ylocal[s] = Usub[s] @ T64^T     (intra-sub-chunk causal conv) -> d_out
// Waves 0-3 compute V, waves 4-7 compute Ylocal (shared A operands in LDS).
// ---------------------------------------------------------------------------
__global__ __launch_bounds__(256) void k_pass1(const float* __restrict__ u,
                                               const float* __restrict__ zt,
                                               const float* __restrict__ tt,
                                               float* __restrict__ V,
                                               float* __restrict__ yout)
{
  __shared__ float Ut[64][68];
  __shared__ float ZTl[64][68];
  __shared__ float TTl[64][68];
  const int tid = threadIdx.x;
  const long s0 = (long)blockIdx.x * 64;
  const float* __restrict__ ubase = u + s0 * 64;
  __builtin_prefetch(ubase + 4096, 0, 1);          // global_prefetch next tile
  for (int idx = tid; idx < 4096; idx += 256) {
    int r = idx >> 6, c = idx & 63;
    Ut[r][c]  = ubase[idx];
    ZTl[r][c] = zt[idx];
    TTl[r][c] = tt[idx];
  }
  __syncthreads();
  const int lane  = tid & 31;
  const int wv    = tid >> 5;
  const int m0    = (wv & 3) * 16;
  const int domT  = wv >> 2;
  const int arow  = m0 + (lane & 15);
  const int khalf = (lane >> 4) * 2;               // A: lanes 0-15 K=0,1; 16-31 K=2,3
  const int crow0 = m0 + (lane >> 4) * 8;          // C: VGPR v -> rows v / v+8
  const float (*Bl)[68] = domT ? TTl : ZTl;
  float* __restrict__ outp = domT ? yout : V;
  for (int nt = 0; nt < 4; ++nt) {
    int ncol = nt * 16 + (lane & 15);
    v8f acc = {0.f, 0.f, 0.f, 0.f, 0.f, 0.f, 0.f, 0.f};
#pragma unroll
    for (int kt = 0; kt < 16; ++kt) {
      int kb = kt * 4 + khalf;
      v2f a, b;
      a.x = Ut[arow][kb];   a.y = Ut[arow][kb + 1];
      b.x = Bl[kb][ncol];   b.y = Bl[kb + 1][ncol];
      acc = wmma4(a, b, acc);
    }
#pragma unroll
    for (int vg = 0; vg < 8; ++vg)
      outp[(s0 + crow0 + vg) * 64 + ncol] = acc[vg];
  }
}

// ---------------------------------------------------------------------------
// Pass 2 (WMMA): batched zero-init group scans. 32 groups per block; state
// X[64 n][32 g] in LDS; each of 64 steps is Xnew = Ab64 @ X + v-slice (WMMA).
// Emits xin0 (entering state per sub-chunk) and group totals e[g].
// ---------------------------------------------------------------------------
__global__ __launch_bounds__(256) void k_scan_groups(const float* __restrict__ V,
                                                     const float* __restrict__ ab64,
                                                     float* __restrict__ xin,
                                                     float* __restrict__ evec)
{
  __shared__ float Al[64][68];
  __shared__ float Xl[64][36];
  const int tid  = threadIdx.x;
  const int gblk = blockIdx.x * 32;
  for (int idx = tid; idx < 4096; idx += 256) Al[idx >> 6][idx & 63] = ab64[idx];
  for (int idx = tid; idx < 64 * 36; idx += 256) (&Xl[0][0])[idx] = 0.0f;
  __syncthreads();
  const int lane  = tid & 31;
  const int wv    = tid >> 5;
  const int n0    = (wv & 3) * 16;
  const int g0    = (wv >> 2) * 16;
  const int arow  = n0 + (lane & 15);
  const int khalf = (lane >> 4) * 2;
  const int crow0 = n0 + (lane >> 4) * 8;
  const int ccol  = g0 + (lane & 15);
  const int sg = tid >> 3, sq = tid & 7;           // store mapping
  for (int i = 0; i < 64; ++i) {
    long sbase = ((long)(gblk + sg) * 64 + i) * 64;
#pragma unroll
    for (int j = 0; j < 8; ++j) xin[sbase + sq * 8 + j] = Xl[sq * 8 + j][sg];
    v8f acc = {0.f, 0.f, 0.f, 0.f, 0.f, 0.f, 0.f, 0.f};
#pragma unroll
    for (int kt = 0; kt < 16; ++kt) {
      int kb = kt * 4 + khalf;
      v2f a, b;
      a.x = Al[arow][kb];  a.y = Al[arow][kb + 1];
      b.x = Xl[kb][ccol];  b.y = Xl[kb + 1][ccol];
      acc = wmma4(a, b, acc);
    }
    float vl[8];
    long vbase = ((long)(gblk + ccol) * 64 + i) * 64;
#pragma unroll
    for (int vg = 0; vg < 8; ++vg) vl[vg] = V[vbase + crow0 + vg];
    __syncthreads();
#pragma unroll
    for (int vg = 0; vg < 8; ++vg) Xl[crow0 + vg][ccol] = acc[vg] + vl[vg];
    __syncthreads();
  }
  long ebase = (long)(gblk + sg) * 64;
#pragma unroll
  for (int j = 0; j < 8; ++j) evec[ebase + sq * 8 + j] = Xl[sq * 8 + j][sg];
}

// ---------------------------------------------------------------------------
// Top scan (single block): X[g] = Ab4096 X[g-1] + e[g]; SBLK[g] = entering state.
// ---------------------------------------------------------------------------
__global__ __launch_bounds__(256) void k_scan_top(const float* __restrict__ ab4096,
                                                  const float* __restrict__ evec,
                                                  float* __restrict__ sblk)
{
  __shared__ float Am[64][65];
  __shared__ float xv[64];
  __shared__ float part[4][64];
  const int tid = threadIdx.x;
  for (int idx = tid; idx < 4096; idx += 256) Am[idx >> 6][idx & 63] = ab4096[idx];
  if (tid < 64) xv[tid] = 0.0f;
  __syncthreads();
  const int n = tid & 63, q = tid >> 6;
  for (int g = 0; g < NGRP; ++g) {
    if (tid < 64) sblk[g * 64 + tid] = xv[tid];
    float s = 0.f;
#pragma unroll
    for (int j = 0; j < 16; ++j) s += Am[n][q * 16 + j] * xv[q * 16 + j];
    part[q][n] = s;
    __syncthreads();
    if (tid < 64)
      xv[tid] = part[0][tid] + part[1][tid] + part[2][tid] + part[3][tid]
              + evec[g * 64 + tid];
    __syncthreads();
  }
}

// ---------------------------------------------------------------------------
// Xin fix (WMMA): Xin[g,i] += Ab64^i @ SBLK[g] == (SBLK @ P64T[i]) rows.
// Block handles one i and 128 groups.
// ---------------------------------------------------------------------------
__global__ __launch_bounds__(256) void k_xin_fix(const float* __restrict__ sblk,
                                                 const float* __restrict__ p64t,
                                                 float* __restrict__ xin)
{
  __shared__ float Sb[128][68];
  __shared__ float Pt[64][68];
  const int tid   = threadIdx.x;
  const int i     = blockIdx.x >> 3;
  const int gbase = (blockIdx.x & 7) * 128;
  for (int idx = tid; idx < 8192; idx += 256)
    Sb[idx >> 6][idx & 63] = sblk[(long)gbase * 64 + idx];
  for (int idx = tid; idx < 4096; idx += 256)
    Pt[idx >> 6][idx & 63] = p64t[(long)i * 4096 + idx];
  __syncthreads();
  const int lane  = tid & 31;
  const int wv    = tid >> 5;
  const int m0    = wv * 16;
  const int arow  = m0 + (lane & 15);
  const int khalf = (lane >> 4) * 2;
  const int crow0 = m0 + (lane >> 4) * 8;
  for (int nt = 0; nt < 4; ++nt) {
    int ncol = nt * 16 + (lane & 15);
    v8f acc = {0.f, 0.f, 0.f, 0.f, 0.f, 0.f, 0.f, 0.f};
#pragma unroll
    for (int kt = 0; kt < 16; ++kt) {
      int kb = kt * 4 + khalf;
      v2f a, b;
      a.x = Sb[arow][kb];  a.y = Sb[arow][kb + 1];
      b.x = Pt[kb][ncol];  b.y = Pt[kb + 1][ncol];
      acc = wmma4(a, b, acc);
    }
#pragma unroll
    for (int vg = 0; vg < 8; ++vg) {
      long g = gbase + crow0 + vg;
      long addr = (g * 64 + i) * 64 + ncol;
      xin[addr] += acc[vg];
    }
  }
}

// ---------------------------------------------------------------------------
// Finalize (WMMA): y = Ylocal + Xin @ R1^T + D*u, C-operand preloaded.
// ---------------------------------------------------------------------------
__global__ __launch_bounds__(256) void k_finalize(const float* __restrict__ u,
                                                  const float* __restrict__ xin,
                                                  const float* __restrict__ r1t,
                                                  const float* __restrict__ Dp,
                                                  float* __restrict__ yout)
{
  __shared__ float Xt[128][68];
  __shared__ float Rl[64][68];
  const int tid = threadIdx.x;
  const long s0 = (long)blockIdx.x * 128;
  for (int idx = tid; idx < 8192; idx += 256)
    Xt[idx >> 6][idx & 63] = xin[s0 * 64 + idx];
  for (int idx = tid; idx < 4096; idx += 256)
    Rl[idx >> 6][idx & 63] = r1t[idx];
  __syncthreads();
  const float Dv  = Dp[0];
  const int lane  = tid & 31;
  const int wv    = tid >> 5;
  const int m0    = wv * 16;
  const int arow  = m0 + (lane & 15);
  const int khalf = (lane >> 4) * 2;
  const int crow0 = m0 + (lane >> 4) * 8;
  for (int nt = 0; nt < 4; ++nt) {
    int ncol = nt * 16 + (lane & 15);
    v8f acc = {0.f, 0.f, 0.f, 0.f, 0.f, 0.f, 0.f, 0.f};
#pragma unroll
    for (int vg = 0; vg < 8; ++vg) {
      long t = (s0 + crow0 + vg) * 64 + ncol;
      acc[vg] = yout[t] + Dv * u[t];               // Ylocal + D*u as C operand
    }
#pragma unroll
    for (int kt = 0; kt < 16; ++kt) {
      int kb = kt * 4 + khalf;
      v2f a, b;
      a.x = Xt[arow][kb];  a.y = Xt[arow][kb + 1];
      b.x = Rl[kb][ncol];  b.y = Rl[kb + 1][ncol];
      acc = wmma4(a, b, acc);
    }
#pragma unroll
    for (int vg = 0; vg < 8; ++vg) {
      long t = (s0 + crow0 + vg) * 64 + ncol;
      yout[t] = acc[vg];
    }
  }
}

// ---------------------------------------------------------------------------
extern "C" void kernel_launch(void* const* d_in, const int* in_sizes, int n_in,
                              void* d_out, int out_size, void* d_ws, size_t ws_size,
                              hipStream_t stream)
{
  const float* u  = (const float*)d_in[0];
  const float* A  = (const float*)d_in[1];
  const float* B  = (const float*)d_in[2];
  const float* C  = (const float*)d_in[3];
  const float* Dp = (const float*)d_in[4];
  const float* ls = (const float*)d_in[5];
  float* ws   = (float*)d_ws;        // needs ~33.6 MiB
  float* yout = (float*)d_out;
  float* V    = ws + OFF_V;
  float* xin  = ws + OFF_XIN;
  float* evec = ws + OFF_E;
  float* sblk = ws + OFF_SBLK;

  k_setup      <<<1,    256, 0, stream>>>(A, B, C, ls, ws);
  k_powers     <<<1,    256, 0, stream>>>(ws);
  k_pass1      <<<NSUB / 64, 256, 0, stream>>>(u, ws + OFF_ZT, ws + OFF_TT, V, yout);
  k_scan_groups<<<NGRP / 32, 256, 0, stream>>>(V, ws + OFF_AB64, xin, evec);
  k_scan_top   <<<1,    256, 0, stream>>>(ws + OFF_AB4096, evec, sblk);
  k_xin_fix    <<<512,  256, 0, stream>>>(sblk, ws + OFF_P64T, xin);
  k_finalize   <<<512,  256, 0, stream>>>(u, xin, ws + OFF_R1T, Dp, yout);
}